// Backprojection_34926674051156
// MI455X (gfx1250) — compile-verified
//
#include <hip/hip_runtime.h>
#include <math.h>

// -----------------------------------------------------------------------------
// 2D parallel-beam backprojection on MI455X (gfx1250), WMMA-based.
//
// vol[z,y,x] = (pi/720) * sum_v lerp(proj[v,z,:], x_c*cos + y_c*sin + ctr)
//   proj[v,z,d] = x[z,0,v,d];  x: (32,1,720,1024) f32;  vol: (32,1,512,512) f32
//
// Per 16x16 pixel tile (one workgroup, 256 thr = 8 waves):
//   * per view: stage a 32(z) x 32(det) zero-filled window into LDS via
//     global_load_async_to_lds_b32 (+ s_wait_asynccnt)
//   * pick line orientation along the axis with |trig| <= sqrt(1/2) so a
//     16-pixel line's 2-tap footprint always fits K=16
//   * lerp weights built as the hat function max(0, 1-|t-k|): 3 VALU/weight,
//     no compares/selects/floor in the inner loop
//   * D[16 px, 16 z] += W[16 px,16 k] x S[16 k,16 z] as 4x v_wmma_f32_16x16x4_f32
//   * two orientation phases accumulate in regs; merged via padded LDS tile
// -----------------------------------------------------------------------------

typedef float v2f __attribute__((ext_vector_type(2)));
typedef float v8f __attribute__((ext_vector_type(8)));

#define NVIEW  720
#define NDCT   1024
#define NIMG   512
#define NZ     32
#define TILE   16
#define SINO_W 32    // staged detector bins per view (union footprint <= 25)
#define SINO_P 35    // LDS row pitch (floats): conflict-free for B-loads
#define VOL_P  257   // LDS z-pitch (floats) for the 16x16x32 volume tile

#if defined(__gfx1250__) && __has_builtin(__builtin_amdgcn_global_load_async_to_lds_b32) && __has_builtin(__builtin_amdgcn_s_wait_asynccnt)
#define HAVE_ASYNC_LDS 1
typedef __attribute__((address_space(1))) int gas_int;   // global
typedef __attribute__((address_space(3))) int las_int;   // LDS
#else
#define HAVE_ASYNC_LDS 0
#endif

__global__ __launch_bounds__(256)
void backproject_wmma(const float* __restrict__ src, float* __restrict__ out)
{
    __shared__ float sino[NZ * SINO_P];     //  4.4 KB  per-view detector window
    __shared__ float voltile[NZ * VOL_P];   // 32.1 KB  16x16x32 output tile

    const int tid   = threadIdx.x;
    const int lane  = tid & 31;
    const int wv    = tid >> 5;        // wave 0..7
    const int zh    = wv & 1;          // z half: z in [16*zh, 16*zh+16)
    const int rbase = (wv >> 1) * 4;   // this wave owns 4 pixel lines
    const int m     = lane & 15;       // lane coord: A-matrix M / B-matrix N
    const int hi    = lane >> 4;       // lane half (selects K sub-pair)

    const int x0 = blockIdx.x * TILE;
    const int y0 = blockIdx.y * TILE;

    const float ctr = (NDCT - 1) * 0.5f;   // 511.5
    const float cx  = (NIMG - 1) * 0.5f;   // 255.5

    // phase 0: theta in [pi/4, 3pi/4)  -> lines along x (|cos| <= sqrt(1/2))
    // phase 1: remaining views         -> lines along y (|sin| <= sqrt(1/2))
    const int sbeg[2][2] = { {180, 540}, {0, 540} };
    const int send[2][2] = { {540, 540}, {180, 720} };

    for (int ph = 0; ph < 2; ++ph) {
        v8f acc[4] = {};   // 4 lines x (16 px x 16 z) f32 accumulators

        for (int seg = 0; seg < 2; ++seg) {
            for (int v = sbeg[ph][seg]; v < send[ph][seg]; ++v) {
                const float theta = (float)v * (float)(M_PI / NVIEW);
                const float c = cosf(theta), s = sinf(theta);

                // detector window origin for the whole 16x16 tile at this view
                const float t00  = ((float)x0 - cx) * c + ((float)y0 - cx) * s + ctr;
                const float tmin = t00 + fminf(15.f * c, 0.f) + fminf(15.f * s, 0.f);
                const int   d_lo = (int)floorf(tmin);

                __syncthreads();   // previous view's LDS reads are done

                // ---- stage sino window: 32 z-rows x 32 bins, zero-filled OOB
                #pragma unroll
                for (int i = 0; i < 4; ++i) {
                    const int e  = tid + 256 * i;       // wave-contiguous rows
                    const int zz = e >> 5, k = e & 31;
                    const int d  = d_lo + k;
                    const bool valid = ((unsigned)d < (unsigned)NDCT);
                    const float* gp = src + ((size_t)zz * (NVIEW * NDCT)
                                             + (size_t)v * NDCT + d);
#if HAVE_ASYNC_LDS
                    if (valid) {
                        __builtin_amdgcn_global_load_async_to_lds_b32(
                            (gas_int*)gp,
                            (las_int*)&sino[zz * SINO_P + k],
                            0, 0);
                    } else {
                        sino[zz * SINO_P + k] = 0.f;
                    }
#else
                    sino[zz * SINO_P + k] = valid ? __ldg(gp) : 0.f;
#endif
                }
#if HAVE_ASYNC_LDS
                __builtin_amdgcn_s_wait_asynccnt(0);
#endif
                __syncthreads();

                const float step = (ph == 0) ? c : s;   // |step| <= sqrt(1/2)

                #pragma unroll
                for (int jj = 0; jj < 4; ++jj) {
                    // t at line start (pixel index m = 0)
                    float rowt0;
                    if (ph == 0)   // line along x at y = y0+rbase+jj
                        rowt0 = ((float)x0 - cx) * c + ((float)(y0 + rbase + jj) - cx) * s + ctr;
                    else           // line along y at x = x0+rbase+jj
                        rowt0 = ((float)(x0 + rbase + jj) - cx) * c + ((float)y0 - cx) * s + ctr;

                    const int rdlo = (int)floorf(rowt0 + fminf(15.f * step, 0.f));
                    const int off  = rdlo - d_lo;             // in [0,16]
                    // this lane's pixel position within the K=16 window
                    const float u  = (rowt0 - (float)rdlo) + step * (float)m;

                    // B rows: lane's z = zh*16 + m  (N coordinate)
                    const float* srow = &sino[(zh * 16 + m) * SINO_P + off];

                    #pragma unroll
                    for (int q = 0; q < 4; ++q) {
                        // A 16x4 f32 layout: lanes 0-15 hold K={4q,4q+1},
                        // lanes 16-31 hold K={4q+2,4q+3} (VGPR0/VGPR1).
                        // Lerp weight at bin k is the hat fn max(0, 1-|u-k|).
                        const int   kq  = q * 4 + hi * 2;
                        const float kqf = (float)kq;
                        const float a0  = fmaxf(0.f, 1.f - fabsf(u - kqf));
                        const float a1  = fmaxf(0.f, 1.f - fabsf(u - kqf - 1.f));
                        v2f A = { a0, a1 };
                        v2f B = { srow[kq], srow[kq + 1] };   // same K pair, this z
                        acc[jj] = __builtin_amdgcn_wmma_f32_16x16x4_f32(
                            false, A, false, B, (short)0, acc[jj], false, false);
                    }
                }
            }
        }

        // ---- dump this phase's accumulators into the LDS volume tile.
        // C/D layout: VGPR r -> lanes 0-15: (M=r,   N=lane),
        //                       lanes16-31: (M=r+8, N=lane-16)
        __syncthreads();
        #pragma unroll
        for (int jj = 0; jj < 4; ++jj) {
            const int row = rbase + jj;
            #pragma unroll
            for (int r = 0; r < 8; ++r) {
                const int M   = r + hi * 8;
                const int xin = (ph == 0) ? M   : row;
                const int yin = (ph == 0) ? row : M;
                const int idx = (zh * 16 + m) * VOL_P + yin * TILE + xin;
                if (ph == 0) voltile[idx]  = acc[jj][r];
                else         voltile[idx] += acc[jj][r];
            }
        }
        __syncthreads();
    }

    // ---- coalesced writeout with the pi/nView weighting
    const float scale = (float)(M_PI / NVIEW);
    #pragma unroll
    for (int i = 0; i < 32; ++i) {
        const int e   = tid + 256 * i;
        const int zz  = e >> 8;
        const int rem = e & 255;
        out[(size_t)zz * (NIMG * NIMG) + (size_t)(y0 + (rem >> 4)) * NIMG
            + (x0 + (rem & 15))] = voltile[zz * VOL_P + rem] * scale;
    }
}

extern "C" void kernel_launch(void* const* d_in, const int* in_sizes, int n_in,
                              void* d_out, int out_size, void* d_ws, size_t ws_size,
                              hipStream_t stream) {
    (void)in_sizes; (void)n_in; (void)out_size; (void)d_ws; (void)ws_size;
    const float* x = (const float*)d_in[0];
    float* out = (float*)d_out;
    dim3 grid(NIMG / TILE, NIMG / TILE, 1);   // 32 x 32 tiles
    backproject_wmma<<<grid, 256, 0, stream>>>(x, out);
}